// GlobalNet_69776038690929
// MI455X (gfx1250) — compile-verified
//
#include <hip/hip_runtime.h>

typedef __attribute__((ext_vector_type(2))) float v2f;
typedef __attribute__((ext_vector_type(8))) float v8f;

#define NPOINTS 8192
#define BATCH   16
#define KSAMP   1024
#define FPS_THREADS 1024
#define PPT (NPOINTS / FPS_THREADS)   // 8 points per thread
#define NPTS 32                       // points per MLP block
#define LSTRIDE 33                    // LDS row stride (bank-conflict padding)
#define NCH_TOTAL 1056                // 32+64+64+128+256+512 folded BN channels

// ---------------------------------------------------------------------------
// All model pointers in one kernarg struct.
// ---------------------------------------------------------------------------
struct Params {
  const float* w[6];
  const float* b[6];
  const float* g[6];
  const float* be[6];
  const float* mn[6];
  const float* vr[6];
};

// ---------------------------------------------------------------------------
// Kernel 0: fold BN(eval)+conv bias into per-channel scale/shift:
//   y = relu((W h + b)*inv + beta - mean*inv),  inv = gamma*rsqrt(var+eps)
//   scale = inv, shift = b*inv + beta - mean*inv
// ---------------------------------------------------------------------------
__global__ __launch_bounds__(256) void bn_fold_kernel(Params prm,
                                                      float* __restrict__ scale,
                                                      float* __restrict__ shift) {
  const int i = blockIdx.x * 256 + threadIdx.x;
  if (i >= NCH_TOTAL) return;
  const int starts[7] = {0, 32, 96, 160, 288, 544, NCH_TOTAL};
  int l = 0;
  while (i >= starts[l + 1]) ++l;
  const int o = i - starts[l];
  float inv = prm.g[l][o] * rsqrtf(prm.vr[l][o] + 1e-5f);
  scale[i] = inv;
  shift[i] = fmaf(-prm.mn[l][o], inv, prm.be[l][o]) + prm.b[l][o] * inv;
}

// ---------------------------------------------------------------------------
// Kernel 1: farthest point sampling, one 1024-thread workgroup per batch.
// Points + running min-distances live in registers. Argmax uses a monotonic
// packed u64 key (dist bits << 32 | ~index): wave32 shuffle butterflies, then
// ds_max_u64 atomics from the 32 wave leaders into a ping-pong LDS slot.
// Only 2 workgroup barriers per sequential step.
// ---------------------------------------------------------------------------
__global__ __launch_bounds__(FPS_THREADS) void fps_kernel(
    const float* __restrict__ xyz, float* __restrict__ sxyz) {
  const int b   = blockIdx.x;
  const int tid = threadIdx.x;
  const float* base = xyz + (size_t)b * NPOINTS * 3;

  float px[PPT], py[PPT], pz[PPT], dd[PPT];
  unsigned lowkey[PPT];
#pragma unroll
  for (int j = 0; j < PPT; ++j) {
    int p = tid + j * FPS_THREADS;
    px[j] = base[p * 3 + 0];
    py[j] = base[p * 3 + 1];
    pz[j] = base[p * 3 + 2];
    dd[j] = 3.4e38f;
    lowkey[j] = ~(unsigned)p;   // larger key == smaller index (tie-break)
  }

  __shared__ float bc[3];
  __shared__ unsigned long long slot[2];
  if (tid == 0) { slot[0] = 0ull; slot[1] = 0ull; }

  int far = 0;  // reference: first sample is index 0
  for (int s = 0; s < KSAMP; ++s) {
    // Owner of current farthest point broadcasts coords + emits sample s.
    if (tid == (far & (FPS_THREADS - 1))) {
      int j = far >> 10;
      bc[0] = px[j]; bc[1] = py[j]; bc[2] = pz[j];
      float* o = sxyz + ((size_t)b * KSAMP + s) * 3;
      o[0] = px[j]; o[1] = py[j]; o[2] = pz[j];
    }
    if (s == KSAMP - 1) break;      // last step needs no further update
    __syncthreads();                // bc (and slot init) visible to all
    const float cx = bc[0], cy = bc[1], cz = bc[2];

    unsigned long long bk = 0ull;
#pragma unroll
    for (int j = 0; j < PPT; ++j) {
      float dx = px[j] - cx, dy = py[j] - cy, dz = pz[j] - cz;
      float d  = fmaf(dx, dx, fmaf(dy, dy, dz * dz));
      dd[j] = fminf(dd[j], d);      // d >= 0 -> float bits are monotonic
      unsigned long long key =
          ((unsigned long long)__float_as_uint(dd[j]) << 32) | lowkey[j];
      bk = (key > bk) ? key : bk;   // ascending j keeps first index on ties
    }
    // wave32 max butterfly on packed keys
#pragma unroll
    for (int off = 16; off > 0; off >>= 1) {
      unsigned long long ok = __shfl_xor(bk, off, 32);
      bk = (ok > bk) ? ok : bk;
    }
    if ((tid & 31) == 0) atomicMax(&slot[s & 1], bk);  // ds_max_u64
    if (tid == 0) slot[(s + 1) & 1] = 0ull;            // reset other slot
    __syncthreads();
    far = (int)~((unsigned)slot[s & 1]);
  }
}

// ---------------------------------------------------------------------------
// Kernel 2: 6-layer pointwise MLP on the 1024 sampled points only, using
// V_WMMA_F32_16X16X4_F32 (D = A x B + C).  A = weight 16x4 (MxK) from global
// (L2-resident), B = activation 4x16 (KxN) from LDS ping-pong buffers.
// ---------------------------------------------------------------------------
template <int CINR, int CINP, int COUT>
__device__ inline void layer(const float* __restrict__ W,
                             const float* __restrict__ scale,
                             const float* __restrict__ shift,
                             const float* __restrict__ src,
                             float* __restrict__ dst,
                             int wave, int lane) {
  constexpr int NT_N   = NPTS / 16;            // 2 point tiles
  constexpr int nTiles = (COUT / 16) * NT_N;
  const int m16  = lane & 15;
  const int koff = (lane < 16) ? 0 : 2;        // A/B frag K split per ISA layout

  for (int t = wave; t < nTiles; t += 8) {
    const int mBase = (t / NT_N) * 16;
    const int nBase = (t % NT_N) * 16;
    const int row = mBase + m16;
    const int col = nBase + m16;
    {
      int tn = t + 8;                          // warm L0 for the next tile
      if (tn < nTiles)
        __builtin_prefetch(&W[((tn / NT_N) * 16 + m16) * CINR], 0, 0);
    }
    v8f acc = {0.f, 0.f, 0.f, 0.f, 0.f, 0.f, 0.f, 0.f};
#pragma unroll 4
    for (int k = 0; k < CINP; k += 4) {
      const int k0 = k + koff, k1 = k0 + 1;
      v2f a, bf;
      a.x  = (k0 < CINR) ? W[row * CINR + k0] : 0.f;   // guard only layer 0 pad
      a.y  = (k1 < CINR) ? W[row * CINR + k1] : 0.f;
      bf.x = src[k0 * LSTRIDE + col];
      bf.y = src[k1 * LSTRIDE + col];
      acc = __builtin_amdgcn_wmma_f32_16x16x4_f32(
          false, a, false, bf, (short)0, acc, false, false);
    }
    // fused pre-folded BN/bias + ReLU on the C tile
    const int rowAdd = (lane < 16) ? 0 : 8;
    const float4 s0 = *(const float4*)(scale + mBase + rowAdd);
    const float4 s1 = *(const float4*)(scale + mBase + rowAdd + 4);
    const float4 h0 = *(const float4*)(shift + mBase + rowAdd);
    const float4 h1 = *(const float4*)(shift + mBase + rowAdd + 4);
    const float sc[8] = {s0.x, s0.y, s0.z, s0.w, s1.x, s1.y, s1.z, s1.w};
    const float sh[8] = {h0.x, h0.y, h0.z, h0.w, h1.x, h1.y, h1.z, h1.w};
#pragma unroll
    for (int r = 0; r < 8; ++r) {
      int o = mBase + r + rowAdd;
      float v = fmaf(acc[r], sc[r], sh[r]);
      dst[o * LSTRIDE + col] = fmaxf(v, 0.f);
    }
  }
}

__global__ __launch_bounds__(256) void mlp_kernel(
    const float* __restrict__ sxyz, Params prm,
    const float* __restrict__ scale, const float* __restrict__ shift,
    float* __restrict__ out) {
  extern __shared__ float smem[];
  float* buf0 = smem;                    // up to 512 rows
  float* buf1 = smem + 512 * LSTRIDE;    // up to 256 rows
  const int b     = blockIdx.x >> 5;     // 32 blocks per batch
  const int pBase = (blockIdx.x & 31) * NPTS;
  const int tid  = threadIdx.x;
  const int wave = tid >> 5, lane = tid & 31;

  // Stage sampled xyz as rows 0..2 of buf0; row 3 zeroed (K padding 3->4).
  if (tid < NPTS * 3) {
    int p = tid / 3, c = tid % 3;
    buf0[c * LSTRIDE + p] = sxyz[((size_t)b * KSAMP + pBase + p) * 3 + c];
  } else if (tid < NPTS * 4) {
    buf0[3 * LSTRIDE + (tid - NPTS * 3)] = 0.f;
  }
  __syncthreads();

  layer<3,   4,   32 >(prm.w[0], scale + 0,   shift + 0,   buf0, buf1, wave, lane);
  __syncthreads();
  layer<32,  32,  64 >(prm.w[1], scale + 32,  shift + 32,  buf1, buf0, wave, lane);
  __syncthreads();
  layer<64,  64,  64 >(prm.w[2], scale + 96,  shift + 96,  buf0, buf1, wave, lane);
  __syncthreads();
  layer<64,  64,  128>(prm.w[3], scale + 160, shift + 160, buf1, buf0, wave, lane);
  __syncthreads();
  layer<128, 128, 256>(prm.w[4], scale + 288, shift + 288, buf0, buf1, wave, lane);
  __syncthreads();
  layer<256, 256, 512>(prm.w[5], scale + 544, shift + 544, buf1, buf0, wave, lane);
  __syncthreads();

  // Transposed writeback: [B, K, 512], coalesced over channels; LDS stride-33
  // padding makes the strided LDS reads bank-conflict-free.
  const size_t obase = ((size_t)b * KSAMP + pBase) * 512;
  for (int i = tid; i < 512 * NPTS; i += 256) {
    int o = i & 511, p = i >> 9;
    out[obase + (size_t)p * 512 + o] = buf0[o * LSTRIDE + p];
  }
}

// ---------------------------------------------------------------------------
extern "C" void kernel_launch(void* const* d_in, const int* in_sizes, int n_in,
                              void* d_out, int out_size, void* d_ws, size_t ws_size,
                              hipStream_t stream) {
  const float* x = (const float*)d_in[0];
  Params prm;
  for (int i = 0; i < 6; ++i) {
    prm.w[i]  = (const float*)d_in[1  + i];
    prm.b[i]  = (const float*)d_in[7  + i];
    prm.g[i]  = (const float*)d_in[13 + i];
    prm.be[i] = (const float*)d_in[19 + i];
    prm.mn[i] = (const float*)d_in[25 + i];
    prm.vr[i] = (const float*)d_in[31 + i];
  }
  // workspace layout: sampled xyz [16,1024,3] | scale[1056] | shift[1056]
  float* sxyz  = (float*)d_ws;
  float* scale = sxyz + (size_t)BATCH * KSAMP * 3;
  float* shift = scale + NCH_TOTAL;

  bn_fold_kernel<<<(NCH_TOTAL + 255) / 256, 256, 0, stream>>>(prm, scale, shift);
  fps_kernel<<<BATCH, FPS_THREADS, 0, stream>>>(x, sxyz);

  const size_t ldsBytes = (size_t)(512 + 256) * LSTRIDE * sizeof(float);  // ~101 KB
  (void)hipFuncSetAttribute((const void*)mlp_kernel,
                            hipFuncAttributeMaxDynamicSharedMemorySize,
                            (int)ldsBytes);
  mlp_kernel<<<BATCH * (KSAMP / NPTS), 256, ldsBytes, stream>>>(
      sxyz, prm, scale, shift, (float*)d_out);
}